// Polar2Cart_64561948393605
// MI455X (gfx1250) — compile-verified
//
#include <hip/hip_runtime.h>
#include <math.h>

// ---- problem geometry (fixed by the reference) ----
#define HP    362
#define WP    362
#define NP    (HP * WP)       // 131044 output pixels per batch
#define AR    512             // angle_res (input dim 1)
#define RR    512             // r_res     (input dim 2)
#define SLICE (AR * RR)       // 262144 elements per batch image (1 MB)

// Replicates the numpy float32 pipeline from _polar_coords/_interp_tables
// for one output pixel p. Produces fused corner offset (f0*RR+f1) and the
// two lerp weights.
__device__ __forceinline__ void polar_tables(int p, int& off, float& a0, float& a1) {
    const int row = p / WP;             // y
    const int col = p - row * WP;       // x
    const float x = (float)col - 181.0f;          // w/2 = 181.0
    const float y = (float)row - 181.0f;          // h/2 = 181.0

    // r = sqrt(x^2+y^2) / sqrt(h^2+w^2)   (all float32, true divide)
    const float denom = sqrtf(262088.0f);         // h*h + w*w = 262088
    float rn = sqrtf(x * x + y * y) / denom;

    const float th0 = atan2f(y, x);

    // sign(cos(th0)) with numpy semantics (exact-zero -> 0)
    const float c = cosf(th0);
    const float sgn = (c > 0.0f) ? 1.0f : ((c < 0.0f) ? -1.0f : 0.0f);

    // r = (r*sign + 0.5) * (r_res-1)
    float r = (rn * sgn + 0.5f) * 511.0f;

    // th = ((atan(tan(th0)) + pi/2) * (angle_res-1)) / pi  -- numpy assoc order
    float th = ((atanf(tanf(th0)) + 1.5707964f) * 511.0f) / 3.1415927f;

    const float f0f = fminf(fmaxf(floorf(th), 0.0f), 510.0f);
    const float f1f = fminf(fmaxf(floorf(r),  0.0f), 510.0f);
    a0 = fminf(fmaxf(th - f0f, 0.0f), 1.0f);
    a1 = fminf(fmaxf(r  - f1f, 0.0f), 1.0f);
    off = (int)f0f * RR + (int)f1f;
}

__device__ __forceinline__ float bilerp(const float* __restrict__ base,
                                        float a0, float a1) {
    const float tl = base[0];
    const float tr = base[1];
    const float bl = base[RR];
    const float br = base[RR + 1];
    const float top = tl + a1 * (tr - tl);
    const float bot = bl + a1 * (br - bl);
    return top + a0 * (bot - top);
}

// Pass 1: batch-invariant interpolation tables -> workspace.
// One 16B entry per pixel: {off(as int bits), a0, a1, pad} so the hot kernel
// fetches the whole entry with a single global_load_b128.
__global__ void p2c_build_tables(float4* __restrict__ tab) {
    const int p = blockIdx.x * blockDim.x + threadIdx.x;
    if (p >= NP) return;
    int off; float a0, a1;
    polar_tables(p, off, a0, a1);
    float4 t;
    t.x = __int_as_float(off);
    t.y = a0;
    t.z = a1;
    t.w = 0.0f;
    tab[p] = t;
}

// Pass 2: gather+lerp. Each thread owns one pixel and 4 consecutive batch
// slices: one b128 table load amortized 4x, per-lane global_prefetch_b8 of
// the next slice's two gather cachelines (slice stride = 1 MB, the whole
// 64 MB input is resident in the 192 MB L2). Output stores are non-temporal
// so the write stream does not evict the L2-resident input.
__global__ void p2c_gather(const float* __restrict__ in,
                           const float4* __restrict__ tab,
                           float* __restrict__ out) {
    const int p = blockIdx.x * blockDim.x + threadIdx.x;
    if (p >= NP) return;
    const int n0 = blockIdx.y * 4;

    const float4 t  = tab[p];            // single global_load_b128
    const int   off = __float_as_int(t.x);
    const float a0  = t.y;
    const float a1  = t.z;

    const float* base = in + (size_t)n0 * SLICE + off;
    float*       o    = out + (size_t)n0 * NP + p;

    #pragma unroll
    for (int i = 0; i < 4; ++i) {
        if (i < 3) {
            // gfx1250 per-lane prefetch of next slice's gather lines
            __builtin_prefetch(base + SLICE,      0, 0);
            __builtin_prefetch(base + SLICE + RR, 0, 0);
        }
        const float v = bilerp(base, a0, a1);
        __builtin_nontemporal_store(v, o);
        base += SLICE;
        o    += NP;
    }
}

// Fallback if the workspace is too small: fused kernel, trig computed once
// per pixel, all N slices looped in-thread with next-slice prefetch.
__global__ void p2c_fused(const float* __restrict__ in,
                          float* __restrict__ out, int N) {
    const int p = blockIdx.x * blockDim.x + threadIdx.x;
    if (p >= NP) return;
    int off; float a0, a1;
    polar_tables(p, off, a0, a1);

    const float* base = in + off;
    float*       o    = out + p;
    for (int n = 0; n < N; ++n) {
        if (n + 1 < N) {
            __builtin_prefetch(base + SLICE,      0, 0);
            __builtin_prefetch(base + SLICE + RR, 0, 0);
        }
        const float v = bilerp(base, a0, a1);
        __builtin_nontemporal_store(v, o);
        base += SLICE;
        o    += NP;
    }
}

extern "C" void kernel_launch(void* const* d_in, const int* in_sizes, int n_in,
                              void* d_out, int out_size, void* d_ws, size_t ws_size,
                              hipStream_t stream) {
    const float* in  = (const float*)d_in[0];
    float*       out = (float*)d_out;
    const int N = in_sizes[0] / SLICE;   // 64 batch slices

    const size_t need = (size_t)NP * sizeof(float4);   // ~2.1 MB
    const int threads = 256;             // 8 waves (wave32) per workgroup
    const int pblocks = (NP + threads - 1) / threads;

    if (ws_size >= need && (N % 4) == 0) {
        float4* tab = (float4*)d_ws;
        p2c_build_tables<<<dim3(pblocks), dim3(threads), 0, stream>>>(tab);
        p2c_gather<<<dim3(pblocks, N / 4), dim3(threads), 0, stream>>>(in, tab, out);
    } else {
        p2c_fused<<<dim3(pblocks), dim3(threads), 0, stream>>>(in, out, N);
    }
}